// AdaBiD_74105365725302
// MI455X (gfx1250) — compile-verified
//
#include <hip/hip_runtime.h>
#include <math.h>

// Problem constants (from reference)
#define Bc     8
#define Tc     12
#define Nc     1536
#define Hc     64
#define HID2c  128
#define TOUTc  12
#define BNc    (Bc * Nc)          // 12288
#define BNTc   (BNc * Tc)         // 147456
#define DELTAc 0.5f
#define KLEPSc 1e-10f
#define BNEPSc 1e-5f
#define DEGMINc 1e-6f

typedef float v2f __attribute__((ext_vector_type(2)));
typedef float v8f __attribute__((ext_vector_type(8)));

// D(16x16) = A(16x4) * B(4x16) + C, fp32 WMMA (CDNA5 V_WMMA_F32_16X16X4_F32)
static __device__ __forceinline__ v8f wmma4(v2f a, v2f b, v8f c) {
    return __builtin_amdgcn_wmma_f32_16x16x4_f32(false, a, false, b, (short)0, c,
                                                 false, false);
}

// ---------------------------------------------------------------------------
// Kernel 1: per (b,n): softmax over T, log-probs, self term, transposed x copy
// x layout (B,T,N,1); outputs P/LP/XT as (b,n,t) contiguous; SELF as (b,n).
// ---------------------------------------------------------------------------
__global__ __launch_bounds__(256) void prep_kernel(const float* __restrict__ x,
                                                   float* __restrict__ P,
                                                   float* __restrict__ LP,
                                                   float* __restrict__ SELF,
                                                   float* __restrict__ XT) {
    int idx = blockIdx.x * 256 + threadIdx.x;
    if (idx >= BNc) return;
    int b = idx / Nc, n = idx % Nc;
    float v[Tc], e[Tc];
    float mx = -3.4e38f;
#pragma unroll
    for (int t = 0; t < Tc; ++t) {
        v[t] = x[((size_t)b * Tc + t) * Nc + n];
        mx = fmaxf(mx, v[t]);
    }
    float s = 0.f;
#pragma unroll
    for (int t = 0; t < Tc; ++t) { e[t] = expf(v[t] - mx); s += e[t]; }
    float inv = 1.f / s;
    float self = 0.f;
#pragma unroll
    for (int t = 0; t < Tc; ++t) {
        float p  = e[t] * inv;
        float lp = logf(p + KLEPSc);
        P[(size_t)idx * Tc + t]  = p;
        LP[(size_t)idx * Tc + t] = lp;
        XT[(size_t)idx * Tc + t] = v[t];
        self += p * lp;
    }
    SELF[idx] = self;
}

// ---------------------------------------------------------------------------
// Kernel 2: fused adjacency construction + aggregation (MODE templated so the
// inner 96-tile loop is branch-free straight-line WMMA code).
//   MODE 0 (fwd): Y[i,t] = sum_j A[i,j]*X[j,t],  Sum[i] = sum_j A[i,j]
//   MODE 1 (bwd): Y[j,t] = sum_i A[i,j]*X[i,t],  Sum[j] = sum_i A[i,j]
// A[i,j] = (self[i] - dot(p_i, logp_j) < DELTA).  A tile made with 3 WMMAs
// (K=12), thresholded, bounced through LDS, consumed by 4 WMMAs with a
// ones-column at t=12 so Sum comes out of the same accumulator.
// ---------------------------------------------------------------------------
template <int MODE>
__global__ __launch_bounds__(128) void agg_kernel(const float* __restrict__ P,
                                                  const float* __restrict__ LP,
                                                  const float* __restrict__ SELF,
                                                  const float* __restrict__ XT,
                                                  float* __restrict__ outY,
                                                  float* __restrict__ outSum) {
    __shared__ float atile[4][16][17];
    const int lane = threadIdx.x & 31;
    const int wv   = threadIdx.x >> 5;
    const int l16  = lane & 15;
    const int koff = (lane < 16) ? 0 : 2;
    const int hi8  = (lane < 16) ? 0 : 8;
    const int b      = blockIdx.y;
    const int myBase = (blockIdx.x * 4 + wv) * 16;

    const float* matA = (MODE == 0) ? P : LP;  // WMMA A rows = my tile
    const float* matB = (MODE == 0) ? LP : P;  // WMMA B rows = other tile
    const float* selfB = SELF + b * Nc;

    // A-operand of the G-tile GEMM (constant over the loop), K = T = 12
    const float* pa = matA + (size_t)(b * Nc + myBase + l16) * Tc;
    v2f aG0 = {pa[0 + koff], pa[1 + koff]};
    v2f aG1 = {pa[4 + koff], pa[5 + koff]};
    v2f aG2 = {pa[8 + koff], pa[9 + koff]};

    float svr[8];
    if (MODE == 0) {
#pragma unroll
        for (int r = 0; r < 8; ++r) svr[r] = selfB[myBase + r + hi8];
    }

    // per-lane constants for the [X | ones | 0] B-operand: clamp the column so
    // every load is in-bounds & unconditional; lane-constant cndmask afterwards
    const bool  colIsX = (l16 < 12);
    const float colFv  = (l16 == 12) ? 1.0f : 0.0f;
    const int   lc     = colIsX ? l16 : 11;

    v8f acc = {0.f, 0.f, 0.f, 0.f, 0.f, 0.f, 0.f, 0.f};
    const float* xbase = XT + (size_t)b * Nc * Tc;

    for (int ot = 0; ot < Nc / 16; ++ot) {
        const int oBase = ot * 16;
        const float* pb = matB + (size_t)(b * Nc + oBase + l16) * Tc;
        v2f bG0 = {pb[0 + koff], pb[1 + koff]};
        v2f bG1 = {pb[4 + koff], pb[5 + koff]};
        v2f bG2 = {pb[8 + koff], pb[9 + koff]};
        v8f g = {0.f, 0.f, 0.f, 0.f, 0.f, 0.f, 0.f, 0.f};
        g = wmma4(aG0, bG0, g);
        g = wmma4(aG1, bG1, g);
        g = wmma4(aG2, bG2, g);

        // threshold -> adjacency tile in LDS ([myRow][otherCol]; MODE 1's
        // C layout is already A^T so the same store works)
        float sc = (MODE == 0) ? 0.f : selfB[oBase + l16];
#pragma unroll
        for (int r = 0; r < 8; ++r) {
            float s  = (MODE == 0) ? svr[r] : sc;
            float av = ((s - g[r]) < DELTAc) ? 1.0f : 0.0f;
            atile[wv][r + hi8][l16] = av;
        }
        __builtin_amdgcn_wave_barrier();
        asm volatile("s_wait_dscnt 0x0" ::: "memory");

        // accumulate Y += A_tile(16x16) @ [X_tile(16x12) | ones | 0], K=16
        const float* xb = xbase + (size_t)oBase * Tc;
#pragma unroll
        for (int s = 0; s < 4; ++s) {
            int k0 = 4 * s + koff;
            v2f aY = {atile[wv][l16][k0], atile[wv][l16][k0 + 1]};
            float x0 = xb[(k0 + 0) * Tc + lc];  // unconditional, in-bounds
            float x1 = xb[(k0 + 1) * Tc + lc];
            v2f bY = {colIsX ? x0 : colFv, colIsX ? x1 : colFv};
            acc = wmma4(aY, bY, acc);
        }
        __builtin_amdgcn_wave_barrier();
    }

#pragma unroll
    for (int r = 0; r < 8; ++r) {
        int row = myBase + r + hi8;
        if (l16 < 12)       outY[(size_t)(b * Nc + row) * Tc + l16] = acc[r];
        else if (l16 == 12) outSum[b * Nc + row] = acc[r];
    }
}

// ---------------------------------------------------------------------------
// Kernel 3: fused graph-combine + encoder MLP + decoder MLP + BN + transpose.
// One block (4 waves) owns 16 (b,n) rows. The 768-wide encoder input row
// z[t*64+h] = relu(x*W1[h] + 3*(0.3*sf+0.7*sb)*W2[h]) is regenerated on the
// fly from 24 per-lane registers (no LDS staging needed for the A-operand).
// All GEMMs are f32 WMMA 16x16x4; layer results ping-pong through LDS.
// ---------------------------------------------------------------------------
__global__ __launch_bounds__(128) void mlp_kernel(
    const float* __restrict__ XT, const float* __restrict__ YF,
    const float* __restrict__ RS, const float* __restrict__ YB,
    const float* __restrict__ CS, const float* __restrict__ W1,
    const float* __restrict__ W2, const float* __restrict__ ew1,
    const float* __restrict__ eb1, const float* __restrict__ ew2,
    const float* __restrict__ eb2, const float* __restrict__ ew3,
    const float* __restrict__ eb3, const float* __restrict__ eproj,
    const float* __restrict__ ebg, const float* __restrict__ ebb,
    const float* __restrict__ ebm, const float* __restrict__ ebv,
    const float* __restrict__ dw1, const float* __restrict__ db1,
    const float* __restrict__ dw2, const float* __restrict__ db2,
    const float* __restrict__ dw3, const float* __restrict__ db3,
    const float* __restrict__ dproj, const float* __restrict__ dbg,
    const float* __restrict__ dbb, const float* __restrict__ dbm,
    const float* __restrict__ dbv, float* __restrict__ out) {
    __shared__ float bufA[16][132];  // h1 / g1
    __shared__ float bufB[16][132];  // h2 / g2
    __shared__ float bufC[16][68];   // x_e
    __shared__ float sW1[64], sW2[64];

    const int lane = threadIdx.x & 31;
    const int wv   = threadIdx.x >> 5;
    const int l16  = lane & 15;
    const int koff = (lane < 16) ? 0 : 2;
    const int hi8  = (lane < 16) ? 0 : 8;
    const int rowBase = blockIdx.x * 16;
    const int b       = rowBase / Nc;

    if (threadIdx.x < 64) {
        sW1[threadIdx.x] = W1[threadIdx.x];
        sW2[threadIdx.x] = W2[threadIdx.x];
    }

    // per-lane row coefficients (A-operand row = rowBase + l16)
    float xr[Tc], cf[Tc];
    {
        int R = rowBase + l16;
        float rs = fmaxf(RS[R], DEGMINc);
        float cs = fmaxf(CS[R], DEGMINc);
#pragma unroll
        for (int t = 0; t < Tc; ++t) {
            xr[t] = XT[(size_t)R * Tc + t];
            float sf = YF[(size_t)R * Tc + t] / rs;
            float sb = YB[(size_t)R * Tc + t] / cs;
            cf[t] = 3.0f * (0.3f * sf + 0.7f * sb);
        }
    }
    __syncthreads();

    // ---- encoder layer 1: h1 = relu(z @ ew1 + eb1), K=768, N=128 ----
    for (int nt = wv; nt < 8; nt += 4) {
        int nG = nt * 16 + l16;
        v8f c = {0.f, 0.f, 0.f, 0.f, 0.f, 0.f, 0.f, 0.f};
#pragma unroll
        for (int t = 0; t < Tc; ++t) {
            float xa = xr[t], ca = cf[t];
            for (int u = 0; u < 16; ++u) {
                int h0 = u * 4 + koff;
                int k0 = t * 64 + h0;
                v2f a = {fmaxf(xa * sW1[h0] + ca * sW2[h0], 0.f),
                         fmaxf(xa * sW1[h0 + 1] + ca * sW2[h0 + 1], 0.f)};
                v2f bb = {ew1[(size_t)k0 * 128 + nG],
                          ew1[(size_t)(k0 + 1) * 128 + nG]};
                c = wmma4(a, bb, c);
            }
        }
#pragma unroll
        for (int r = 0; r < 8; ++r)
            bufA[r + hi8][nG] = fmaxf(c[r] + eb1[nG], 0.f);
    }
    __syncthreads();

    // ---- encoder layer 2: h2 = relu(h1 @ ew2 + eb2), K=128, N=128 ----
    for (int nt = wv; nt < 8; nt += 4) {
        int nG = nt * 16 + l16;
        v8f c = {0.f, 0.f, 0.f, 0.f, 0.f, 0.f, 0.f, 0.f};
        for (int s = 0; s < 32; ++s) {
            int k0 = 4 * s + koff;
            v2f a  = {bufA[l16][k0], bufA[l16][k0 + 1]};
            v2f bb = {ew2[(size_t)k0 * 128 + nG],
                      ew2[(size_t)(k0 + 1) * 128 + nG]};
            c = wmma4(a, bb, c);
        }
#pragma unroll
        for (int r = 0; r < 8; ++r)
            bufB[r + hi8][nG] = fmaxf(c[r] + eb2[nG], 0.f);
    }
    __syncthreads();

    // ---- encoder layer 3 + proj + BN: x_e, N=64 (one n-tile per wave) ----
    {
        int nG = wv * 16 + l16;
        v8f c = {0.f, 0.f, 0.f, 0.f, 0.f, 0.f, 0.f, 0.f};
        for (int s = 0; s < 32; ++s) {  // h2 @ ew3, K=128
            int k0 = 4 * s + koff;
            v2f a  = {bufB[l16][k0], bufB[l16][k0 + 1]};
            v2f bb = {ew3[(size_t)k0 * 64 + nG],
                      ew3[(size_t)(k0 + 1) * 64 + nG]};
            c = wmma4(a, bb, c);
        }
#pragma unroll
        for (int t = 0; t < Tc; ++t) {  // z @ eproj, K=768
            float xa = xr[t], ca = cf[t];
            for (int u = 0; u < 16; ++u) {
                int h0 = u * 4 + koff;
                int k0 = t * 64 + h0;
                v2f a = {fmaxf(xa * sW1[h0] + ca * sW2[h0], 0.f),
                         fmaxf(xa * sW1[h0 + 1] + ca * sW2[h0 + 1], 0.f)};
                v2f bb = {eproj[(size_t)k0 * 64 + nG],
                          eproj[(size_t)(k0 + 1) * 64 + nG]};
                c = wmma4(a, bb, c);
            }
        }
        float inv = 1.f / sqrtf(ebv[nG] + BNEPSc);
        float gg = ebg[nG], be = ebb[nG], mm = ebm[nG], b3 = eb3[nG];
#pragma unroll
        for (int r = 0; r < 8; ++r)
            bufC[r + hi8][nG] = (c[r] + b3 - mm) * inv * gg + be;
    }
    __syncthreads();

    // ---- decoder layer 1: g1 = relu(x_e @ dw1 + db1), K=64, N=128 ----
    for (int nt = wv; nt < 8; nt += 4) {
        int nG = nt * 16 + l16;
        v8f c = {0.f, 0.f, 0.f, 0.f, 0.f, 0.f, 0.f, 0.f};
        for (int s = 0; s < 16; ++s) {
            int k0 = 4 * s + koff;
            v2f a  = {bufC[l16][k0], bufC[l16][k0 + 1]};
            v2f bb = {dw1[(size_t)k0 * 128 + nG],
                      dw1[(size_t)(k0 + 1) * 128 + nG]};
            c = wmma4(a, bb, c);
        }
#pragma unroll
        for (int r = 0; r < 8; ++r)
            bufA[r + hi8][nG] = fmaxf(c[r] + db1[nG], 0.f);
    }
    __syncthreads();

    // ---- decoder layer 2: g2 = relu(g1 @ dw2 + db2), K=128, N=128 ----
    for (int nt = wv; nt < 8; nt += 4) {
        int nG = nt * 16 + l16;
        v8f c = {0.f, 0.f, 0.f, 0.f, 0.f, 0.f, 0.f, 0.f};
        for (int s = 0; s < 32; ++s) {
            int k0 = 4 * s + koff;
            v2f a  = {bufA[l16][k0], bufA[l16][k0 + 1]};
            v2f bb = {dw2[(size_t)k0 * 128 + nG],
                      dw2[(size_t)(k0 + 1) * 128 + nG]};
            c = wmma4(a, bb, c);
        }
#pragma unroll
        for (int r = 0; r < 8; ++r)
            bufB[r + hi8][nG] = fmaxf(c[r] + db2[nG], 0.f);
    }
    __syncthreads();

    // ---- decoder layer 3 + proj + BN + transposed store, N=12 (wave 0) ----
    if (wv == 0) {
        // clamp column for unconditional in-bounds weight loads; lane-constant
        // select zeroes the padded columns
        const bool colOk = (l16 < 12);
        const int  ncl   = colOk ? l16 : 11;
        v8f c = {0.f, 0.f, 0.f, 0.f, 0.f, 0.f, 0.f, 0.f};
        for (int s = 0; s < 32; ++s) {  // g2 @ dw3, K=128
            int k0 = 4 * s + koff;
            v2f a = {bufB[l16][k0], bufB[l16][k0 + 1]};
            float w0 = dw3[(size_t)k0 * 12 + ncl];
            float w1 = dw3[(size_t)(k0 + 1) * 12 + ncl];
            v2f bb = {colOk ? w0 : 0.f, colOk ? w1 : 0.f};
            c = wmma4(a, bb, c);
        }
        for (int s = 0; s < 16; ++s) {  // x_e @ dproj, K=64
            int k0 = 4 * s + koff;
            v2f a = {bufC[l16][k0], bufC[l16][k0 + 1]};
            float w0 = dproj[(size_t)k0 * 12 + ncl];
            float w1 = dproj[(size_t)(k0 + 1) * 12 + ncl];
            v2f bb = {colOk ? w0 : 0.f, colOk ? w1 : 0.f};
            c = wmma4(a, bb, c);
        }
        if (colOk) {
            float inv = 1.f / sqrtf(dbv[l16] + BNEPSc);
            float gg = dbg[l16], be = dbb[l16], mm = dbm[l16], b3 = db3[l16];
#pragma unroll
            for (int r = 0; r < 8; ++r) {
                int R = rowBase + r + hi8;
                int n = R % Nc;
                float res = (c[r] + b3 - mm) * inv * gg + be;
                out[((size_t)b * TOUTc + l16) * Nc + n] = res;
            }
        }
    }
}

// ---------------------------------------------------------------------------
extern "C" void kernel_launch(void* const* d_in, const int* in_sizes, int n_in,
                              void* d_out, int out_size, void* d_ws,
                              size_t ws_size, hipStream_t stream) {
    const float* x     = (const float*)d_in[0];
    const float* W1    = (const float*)d_in[1];
    const float* W2    = (const float*)d_in[2];
    const float* ew1   = (const float*)d_in[3];
    const float* eb1   = (const float*)d_in[4];
    const float* ew2   = (const float*)d_in[5];
    const float* eb2   = (const float*)d_in[6];
    const float* ew3   = (const float*)d_in[7];
    const float* eb3   = (const float*)d_in[8];
    const float* eproj = (const float*)d_in[9];
    const float* ebg   = (const float*)d_in[10];
    const float* ebb   = (const float*)d_in[11];
    const float* ebm   = (const float*)d_in[12];
    const float* ebv   = (const float*)d_in[13];
    const float* dw1   = (const float*)d_in[14];
    const float* db1   = (const float*)d_in[15];
    const float* dw2   = (const float*)d_in[16];
    const float* db2   = (const float*)d_in[17];
    const float* dw3   = (const float*)d_in[18];
    const float* db3   = (const float*)d_in[19];
    const float* dproj = (const float*)d_in[20];
    const float* dbg   = (const float*)d_in[21];
    const float* dbb   = (const float*)d_in[22];
    const float* dbm   = (const float*)d_in[23];
    const float* dbv   = (const float*)d_in[24];

    float* ws   = (float*)d_ws;
    float* P    = ws;                      // B*N*T
    float* LP   = ws + BNTc;               // B*N*T
    float* SELF = ws + 2 * BNTc;           // B*N
    float* XT   = ws + 2 * BNTc + BNc;     // B*N*T
    float* YF   = ws + 3 * BNTc + BNc;     // B*N*T
    float* RS   = ws + 4 * BNTc + BNc;     // B*N
    float* YB   = ws + 4 * BNTc + 2 * BNc; // B*N*T
    float* CS   = ws + 5 * BNTc + 2 * BNc; // B*N

    prep_kernel<<<BNc / 256, 256, 0, stream>>>(x, P, LP, SELF, XT);
    dim3 ag(Nc / 16 / 4, Bc);
    agg_kernel<0><<<ag, 128, 0, stream>>>(P, LP, SELF, XT, YF, RS);
    agg_kernel<1><<<ag, 128, 0, stream>>>(P, LP, SELF, XT, YB, CS);
    mlp_kernel<<<BNc / 16, 128, 0, stream>>>(
        XT, YF, RS, YB, CS, W1, W2, ew1, eb1, ew2, eb2, ew3, eb3, eproj, ebg,
        ebb, ebm, ebv, dw1, db1, dw2, db2, dw3, db3, dproj, dbg, dbb, dbm,
        dbv, (float*)d_out);
}